// MixedAttention_27650999451869
// MI455X (gfx1250) — compile-verified
//
#include <hip/hip_runtime.h>

// ---------------------------------------------------------------------------
// MixedAttention (sliding-window GQA) for MI455X / gfx1250.
// - All matrix math via V_WMMA_F32_16X16X4_F32 (exact f32 vs f32 reference).
// - GEMM tiles staged into LDS with GLOBAL_LOAD_ASYNC_TO_LDS_B128 (ASYNCcnt)
//   when the toolchain exposes the builtin; synchronous fallback otherwise.
// ---------------------------------------------------------------------------

#define HEADS_ 16
#define KVH_   4
#define HD_    64
#define HID_   1024
#define NB_    2
#define NS_    2048
#define WIN_   64

typedef float v2f __attribute__((ext_vector_type(2)));
typedef float v4f __attribute__((ext_vector_type(4)));
typedef float v8f __attribute__((ext_vector_type(8)));

#if defined(__AMDGCN__) && __has_builtin(__builtin_amdgcn_global_load_async_to_lds_b128)
#define USE_ASYNC_LDS 1
typedef int i4v __attribute__((__vector_size__(16)));
typedef __attribute__((address_space(1))) i4v gas_i4;   // global int4
typedef __attribute__((address_space(3))) i4v las_i4;   // LDS int4
// Global generic pointer -> AS1 pointer (same 64-bit address).
static __device__ __forceinline__ gas_i4* to_global_i4(const void* p) {
  return (gas_i4*)(unsigned long long)p;
}
// Generic LDS pointer -> AS3 pointer (low 32 bits of generic addr = LDS offset).
static __device__ __forceinline__ las_i4* to_lds_i4(void* p) {
  return (las_i4*)(unsigned int)(unsigned long long)p;
}
#else
#define USE_ASYNC_LDS 0
#endif

static __device__ __forceinline__ void wait_async0() {
#if defined(__AMDGCN__)
#if __has_builtin(__builtin_amdgcn_s_wait_asynccnt)
  __builtin_amdgcn_s_wait_asynccnt(0);
#else
  asm volatile("s_wait_asynccnt 0" ::: "memory");
#endif
#endif
}

static __device__ __forceinline__ v8f wmma4(v2f a, v2f b, v8f c) {
  // D(16x16,f32) += A(16x4,f32) * B(4x16,f32)
  return __builtin_amdgcn_wmma_f32_16x16x4_f32(
      /*neg_a=*/false, a, /*neg_b=*/false, b,
      /*c_mod=*/(short)0, c, /*reuse_a=*/false, /*reuse_b=*/false);
}

static __device__ __forceinline__ v8f zero8() {
  v8f z = {0.f, 0.f, 0.f, 0.f, 0.f, 0.f, 0.f, 0.f};
  return z;
}

// RoPE (interleaved pairs). Pair partner lives on lane^1 by construction:
// within a 16-lane group, column (= head-dim position) follows lane%16.
static __device__ __forceinline__ float rope_apply(float v, int d, int pos) {
  float partner = __shfl_xor(v, 1, 32);
  int t = (d & 63) >> 1;
  // freq = THETA^(-t/32) = 2^(-t * log2(10000)/32)
  float freq = exp2f(-(float)t * 0.41524101186092035f);
  float ang = (float)pos * freq;
  float sn, cs;
  sincosf(ang, &sn, &cs);
  if ((d & 1) == 0) return v * cs - partner * sn;   // r1 = x1*cos - x2*sin
  else              return partner * sn + v * cs;   // r2 = x1*sin + x2*cos
}

// ---------------------------------------------------------------------------
// Shared GEMM tile: block = 128 threads (4 waves). Computes C(16x64) of
// A(M,K=1024) @ W(K=1024,N=ldw) at (m0, n0); wave w owns N sub-tile w.
// A staged 16x32 f32 in LDS (row-major); W staged 32x64 f32 (row-major).
// Staging uses async global->LDS b128 copies when available.
// ---------------------------------------------------------------------------
static __device__ __forceinline__ v8f gemm16x64_tile(
    const float* __restrict__ A, int lda,
    const float* __restrict__ W, int ldw,
    int m0, int n0, float* As, float* Bs) {
  const int tid  = threadIdx.x;      // 0..127
  const int wave = tid >> 5;         // 0..3
  const int lane = tid & 31;
  const int lm   = lane & 15;        // M (A-frag) / N (B-frag) index
  const int lh   = lane >> 4;        // K-half selector

  v8f c = zero8();

  const int arow = tid >> 3;         // 0..15  (A tile row)
  const int akq  = (tid & 7) * 4;    // 0,4,...,28 (A tile K chunk)
  const int bnq  = (tid & 3) * 16;   // B chunk base column per thread

  for (int kb = 0; kb < HID_; kb += 32) {
    __syncthreads();  // protect previous iteration's fragment reads (dscnt+barrier)

#if USE_ASYNC_LDS
    // --- async global -> LDS staging (ASYNCcnt) ---------------------------
    __builtin_amdgcn_global_load_async_to_lds_b128(
        to_global_i4(A + (size_t)(m0 + arow) * lda + kb + akq),
        to_lds_i4(&As[arow * 32 + akq]), 0, 0);
    // B tile: 32 rows x 64 cols = 512 x 16B chunks; 4 chunks per thread.
#pragma unroll
    for (int t = 0; t < 4; ++t) {
      int ch   = tid + 128 * t;        // 0..511
      int krow = ch >> 4;              // 0..31
      int nq   = (ch & 15) * 4;        // 0..60
      __builtin_amdgcn_global_load_async_to_lds_b128(
          to_global_i4(W + (size_t)(kb + krow) * ldw + n0 + nq),
          to_lds_i4(&Bs[krow * 64 + nq]), 0, 0);
    }
    wait_async0();
#else
    // --- synchronous fallback --------------------------------------------
    {   // A tile: 16 rows x 32 K, one float4 per thread
      const float* src = A + (size_t)(m0 + arow) * lda + kb + akq;
      *(v4f*)&As[arow * 32 + akq] = *(const v4f*)src;
    }
    {   // W tile: 32 K x 64 N, 16 floats per thread, contiguous
      int krow = tid >> 2;             // 0..31
      const float* src = W + (size_t)(kb + krow) * ldw + n0 + bnq;
      float* dst = &Bs[krow * 64 + bnq];
#pragma unroll
      for (int j = 0; j < 4; ++j)
        *(v4f*)(dst + 4 * j) = *(const v4f*)(src + 4 * j);
    }
#endif
    __syncthreads();

#pragma unroll
    for (int s = 0; s < 8; ++s) {
      int k0 = s * 4 + 2 * lh;                       // lane's K base
      v2f a = *(const v2f*)&As[lm * 32 + k0];        // A[m][k0], A[m][k0+1]
      v2f b;
      int nn = wave * 16 + lm;
      b.x = Bs[k0 * 64 + nn];                        // B[k0]  [n]
      b.y = Bs[(k0 + 1) * 64 + nn];                  // B[k0+1][n]
      c = wmma4(a, b, c);
    }
  }
  return c;
}

// ---------------------------------------------------------------------------
// Kernel 1: fused Q + KV projection, bias, RoPE, layout permutation.
// grid = (256, 24): y<16 -> Q tile, y>=16 -> KV tile.
// Qr: (B,16,S,64)  Kr,Vr: (B,4,S,64)
// ---------------------------------------------------------------------------
__global__ __launch_bounds__(128) void qkv_proj_kernel(
    const float* __restrict__ hidden,
    const float* __restrict__ q_w, const float* __restrict__ q_b,
    const float* __restrict__ kv_w, const float* __restrict__ kv_b,
    float* __restrict__ Qr, float* __restrict__ Kr, float* __restrict__ Vr) {
  __shared__ __align__(16) float As[16 * 32];
  __shared__ __align__(16) float Bs[32 * 64];

  const int m0  = blockIdx.x * 16;
  const int nt  = blockIdx.y;
  const bool isQ = nt < 16;
  const float* W  = isQ ? q_w : kv_w;
  const float* bb = isQ ? q_b : kv_b;
  const int ldw = isQ ? HID_ : 2 * KVH_ * HD_;     // 1024 or 512
  const int n0  = (isQ ? nt : nt - 16) * 64;

  v8f c = gemm16x64_tile(hidden, HID_, W, ldw, m0, n0, As, Bs);

  const int lane = threadIdx.x & 31;
  const int wave = threadIdx.x >> 5;
  const int lm = lane & 15, lh = lane >> 4;
  const int n = n0 + wave * 16 + lm;               // global output column
  const float bias = bb[n];

#pragma unroll
  for (int r = 0; r < 8; ++r) {
    int m = m0 + r + 8 * lh;                       // global row (b*S + s)
    int bpos = m >> 11;                            // / 2048
    int spos = m & (NS_ - 1);
    float v = c[r] + bias;
    if (isQ) {
      int head = n >> 6, d = n & 63;
      float vr = rope_apply(v, d, spos);
      Qr[(((size_t)bpos * HEADS_ + head) * NS_ + spos) * HD_ + d] = vr;
    } else {
      int kvsel = n >> 8;                          // 0 = K, 1 = V (uniform per wave)
      int head = (n >> 6) & 3, d = n & 63;
      if (kvsel == 0) {
        float vr = rope_apply(v, d, spos);
        Kr[(((size_t)bpos * KVH_ + head) * NS_ + spos) * HD_ + d] = vr;
      } else {
        Vr[(((size_t)bpos * KVH_ + head) * NS_ + spos) * HD_ + d] = v;
      }
    }
  }
}

// ---------------------------------------------------------------------------
// Kernel 2: sliding-window attention. One wave per (b, head, 16-query tile).
// Window [i-64, i+64) for query i -> 9 key tiles of 16 per query tile.
// ---------------------------------------------------------------------------
__global__ __launch_bounds__(32) void attn_kernel(
    const float* __restrict__ Qr, const float* __restrict__ Kr,
    const float* __restrict__ Vr, const float* __restrict__ mask,
    float* __restrict__ Aout) {
  __shared__ __align__(16) float sc[16 * 144];     // scores / probs
  __shared__ float red[32];
  __shared__ float rowsum[16];

  const int qt = blockIdx.x;           // 0..127
  const int h  = blockIdx.y;           // 0..15
  const int b  = blockIdx.z;           // 0..1
  const int i0 = qt * 16;
  const int lane = threadIdx.x;
  const int lm = lane & 15, lh = lane >> 4;
  const int kvh = h >> 2;              // GQA: 4 query heads per KV head

  const float* Qbase = Qr + (((size_t)b * HEADS_ + h) * NS_ + i0) * HD_;
  const float* Kbase = Kr + ((size_t)b * KVH_ + kvh) * NS_ * HD_;
  const float* Vbase = Vr + ((size_t)b * KVH_ + kvh) * NS_ * HD_;

  // Q fragments for all 16 K-steps (head dim 64), pre-scaled by 1/sqrt(64).
  v2f qa[16];
#pragma unroll
  for (int s = 0; s < 16; ++s) {
    int d = s * 4 + 2 * lh;
    v2f t = *(const v2f*)(Qbase + (size_t)lm * HD_ + d);
    qa[s].x = t.x * 0.125f;
    qa[s].y = t.y * 0.125f;
  }

  // --- scores: Q @ K^T, one 16x16 tile per key tile -----------------------
  for (int kt = 0; kt < 9; ++kt) {
    int kb = i0 - WIN_ + kt * 16;
    int key = kb + lm;
    int keyc = key < 0 ? 0 : (key > NS_ - 1 ? NS_ - 1 : key);   // safe address
    const float* Kp = Kbase + (size_t)keyc * HD_;
    v8f c = zero8();
#pragma unroll
    for (int s = 0; s < 16; ++s) {
      int d = s * 4 + 2 * lh;
      v2f kf = *(const v2f*)(Kp + d);
      c = wmma4(qa[s], kf, c);
    }
#pragma unroll
    for (int r = 0; r < 8; ++r) {
      int m = r + 8 * lh;
      int i = i0 + m;
      int j = kb + lm;
      bool valid = (j >= 0) && (j < NS_) && (j >= i - WIN_) && (j < i + WIN_);
      float sv = valid ? (c[r] + mask[(size_t)b * NS_ + j]) : -__builtin_inff();
      sc[m * 144 + kt * 16 + lm] = sv;
    }
  }
  __syncthreads();

  // --- softmax over 144 columns per row (2 lanes per row) -----------------
  {
    int row = lm, half = lh;
    float mx = -__builtin_inff();
    for (int cj = 0; cj < 72; ++cj)
      mx = fmaxf(mx, sc[row * 144 + half * 72 + cj]);
    red[half * 16 + row] = mx;
    __syncthreads();
    float rm = fmaxf(red[row], red[16 + row]);
    float sum = 0.f;
    for (int cj = 0; cj < 72; ++cj) {
      int idx = row * 144 + half * 72 + cj;
      float e = __expf(sc[idx] - rm);
      sc[idx] = e;
      sum += e;
    }
    __syncthreads();
    red[half * 16 + row] = sum;
    __syncthreads();
    if (lane < 16) rowsum[lane] = red[lane] + red[16 + lane];
    __syncthreads();
  }

  // --- P @ V: 36 K-steps over 144 keys, 4 d-tiles of 16 -------------------
  v8f oacc[4];
#pragma unroll
  for (int dt = 0; dt < 4; ++dt) oacc[dt] = zero8();

  for (int s = 0; s < 36; ++s) {
    int kl = s * 4 + 2 * lh;                       // 0..142, even
    v2f pa = *(const v2f*)&sc[lm * 144 + kl];      // P[m][kl], P[m][kl+1]
    int key0 = i0 - WIN_ + kl;
    int key1 = key0 + 1;
    int k0c = key0 < 0 ? 0 : (key0 > NS_ - 1 ? NS_ - 1 : key0);
    int k1c = key1 < 0 ? 0 : (key1 > NS_ - 1 ? NS_ - 1 : key1);
#pragma unroll
    for (int dt = 0; dt < 4; ++dt) {
      int d = dt * 16 + lm;
      v2f vb;
      vb.x = Vbase[(size_t)k0c * HD_ + d];         // prob==0 for clamped keys
      vb.y = Vbase[(size_t)k1c * HD_ + d];
      oacc[dt] = wmma4(pa, vb, oacc[dt]);
    }
  }

  // --- normalize + write (B,S,HID) for the O projection -------------------
#pragma unroll
  for (int dt = 0; dt < 4; ++dt) {
#pragma unroll
    for (int r = 0; r < 8; ++r) {
      int m = r + 8 * lh;
      float o = oacc[dt][r] / rowsum[m];
      int srow = i0 + m;
      Aout[((size_t)b * NS_ + srow) * HID_ + h * HD_ + dt * 16 + lm] = o;
    }
  }
}

// ---------------------------------------------------------------------------
// Kernel 3: output projection, (4096,1024)@(1024,1024)+o_b -> d_out.
// ---------------------------------------------------------------------------
__global__ __launch_bounds__(128) void o_proj_kernel(
    const float* __restrict__ Ain, const float* __restrict__ o_w,
    const float* __restrict__ o_b, float* __restrict__ out) {
  __shared__ __align__(16) float As[16 * 32];
  __shared__ __align__(16) float Bs[32 * 64];
  const int m0 = blockIdx.x * 16;
  const int n0 = blockIdx.y * 64;

  v8f c = gemm16x64_tile(Ain, HID_, o_w, HID_, m0, n0, As, Bs);

  const int lane = threadIdx.x & 31;
  const int wave = threadIdx.x >> 5;
  const int n = n0 + wave * 16 + (lane & 15);
  const float bias = o_b[n];
#pragma unroll
  for (int r = 0; r < 8; ++r) {
    int m = m0 + r + 8 * (lane >> 4);
    out[(size_t)m * HID_ + n] = c[r] + bias;
  }
}

// ---------------------------------------------------------------------------
extern "C" void kernel_launch(void* const* d_in, const int* in_sizes, int n_in,
                              void* d_out, int out_size, void* d_ws, size_t ws_size,
                              hipStream_t stream) {
  (void)in_sizes; (void)n_in; (void)out_size; (void)ws_size;
  const float* hidden = (const float*)d_in[0];
  const float* mask   = (const float*)d_in[1];
  const float* q_w    = (const float*)d_in[2];
  const float* q_b    = (const float*)d_in[3];
  const float* kv_w   = (const float*)d_in[4];
  const float* kv_b   = (const float*)d_in[5];
  const float* o_w    = (const float*)d_in[6];
  const float* o_b    = (const float*)d_in[7];
  float* out = (float*)d_out;

  // Workspace layout (f32):
  //   Qr   : B*16*S*64 = 4M elems
  //   Kr   : B*4*S*64  = 1M elems
  //   Vr   : B*4*S*64  = 1M elems
  //   Aattn: B*S*1024  = 4M elems      (total 40 MB)
  float* Qr    = (float*)d_ws;
  float* Kr    = Qr + (size_t)NB_ * HEADS_ * NS_ * HD_;
  float* Vr    = Kr + (size_t)NB_ * KVH_ * NS_ * HD_;
  float* Aattn = Vr + (size_t)NB_ * KVH_ * NS_ * HD_;

  {
    dim3 g(256, 24), blk(128);   // y<16: Q tiles, y>=16: KV tiles
    qkv_proj_kernel<<<g, blk, 0, stream>>>(hidden, q_w, q_b, kv_w, kv_b, Qr, Kr, Vr);
  }
  {
    dim3 g(NS_ / 16, HEADS_, NB_), blk(32);
    attn_kernel<<<g, blk, 0, stream>>>(Qr, Kr, Vr, mask, Aattn);
  }
  {
    dim3 g(256, 16), blk(128);
    o_proj_kernel<<<g, blk, 0, stream>>>(Aattn, o_w, o_b, out);
  }
}